// RationalQuadraticSpline_52810917871888
// MI455X (gfx1250) — compile-verified
//
#include <hip/hip_runtime.h>
#include <math.h>

#define NBINS  16
#define BOUNDF 5.0f
#define LUTN   1024   // cell width 10/1024 = 0.009766 < min bin width 0.01 -> <=1 knot per cell

typedef __attribute__((ext_vector_type(4))) float v4f;   // clang ext-vector: valid for nontemporal builtins

__global__ __launch_bounds__(256) void rqs_kernel(
    const float* __restrict__ x,
    const float* __restrict__ uw,
    const float* __restrict__ uh,
    const float* __restrict__ ud,
    float* __restrict__ out_y,
    float* __restrict__ out_ld,
    int n4, int n)
{
    __shared__ float4 sP0[NBINS];      // {x_k, 1/w_k, y_k, h_k}  -> 64 dwords = 64 banks, conflict-free
    __shared__ float4 sP1[NBINS];      // {d_k, d_k1, s_k, d_k+d_k1-2s_k}
    __shared__ float  sB[NBINS];       // sB[j] = knot_x[j+1], sB[15] = +inf
    __shared__ int2   sCell[LUTN];     // {.x = bits(next_boundary), .y = bin at cell left edge}

    // ---- per-block spline prologue (tiny: 16 bins; auto-scalarized by compiler) ----
    if (threadIdx.x == 0) {
        float w[NBINS], h[NBINS], d[NBINS + 1];
        float m = uw[0];
        for (int k = 1; k < NBINS; ++k) m = fmaxf(m, uw[k]);
        float s = 0.0f;
        for (int k = 0; k < NBINS; ++k) { w[k] = __expf(uw[k] - m); s += w[k]; }
        float inv = 1.0f / s;
        for (int k = 0; k < NBINS; ++k)
            w[k] = (0.001f + (1.0f - 0.001f * NBINS) * (w[k] * inv)) * (2.0f * BOUNDF);
        m = uh[0];
        for (int k = 1; k < NBINS; ++k) m = fmaxf(m, uh[k]);
        s = 0.0f;
        for (int k = 0; k < NBINS; ++k) { h[k] = __expf(uh[k] - m); s += h[k]; }
        inv = 1.0f / s;
        for (int k = 0; k < NBINS; ++k)
            h[k] = (0.001f + (1.0f - 0.001f * NBINS) * (h[k] * inv)) * (2.0f * BOUNDF);
        for (int k = 0; k <= NBINS; ++k) {
            float v = ud[k];
            d[k] = fmaxf(v, 0.0f) + log1pf(__expf(-fabsf(v))) + 0.001f;  // stable softplus + MIN_DERIVATIVE
        }
        float kx = -BOUNDF, ky = -BOUNDF;
        for (int k = 0; k < NBINS; ++k) {
            float iw = 1.0f / w[k];
            float sk = h[k] * iw;
            sP0[k] = make_float4(kx, iw, ky, h[k]);
            sP1[k] = make_float4(d[k], d[k + 1], sk, d[k] + d[k + 1] - 2.0f * sk);
            kx += w[k];
            ky += h[k];
            if (k < NBINS - 1) sB[k] = kx;   // knot_x[k+1]
        }
        sB[NBINS - 1] = __builtin_inff();    // bin 15 never advances
    }
    __syncthreads();

    // ---- block-parallel fused cell-table build (exact: <=1 boundary per cell) ----
    {
        float b[NBINS - 1];
        #pragma unroll
        for (int j = 0; j < NBINS - 1; ++j) b[j] = sB[j];
        const float cw = (2.0f * BOUNDF) / (float)LUTN;
        for (int i = threadIdx.x; i < LUTN; i += blockDim.x) {
            float c = fmaf((float)i, cw, -BOUNDF);
            int bin = 0;
            #pragma unroll
            for (int j = 0; j < NBINS - 1; ++j) bin += (b[j] < c) ? 1 : 0;
            float nb = sB[bin];                       // next boundary after this bin
            sCell[i] = make_int2(__float_as_int(nb), bin);
        }
    }
    __syncthreads();

    const float cellScale = (float)LUTN / (2.0f * BOUNDF);   // 102.4
    const float cellBias  = cellScale * BOUNDF;               // 512.0

    const v4f* __restrict__ x4 = (const v4f*)x;
    v4f* __restrict__ y4 = (v4f*)out_y;
    v4f* __restrict__ l4 = (v4f*)out_ld;

    const int tid    = blockIdx.x * blockDim.x + threadIdx.x;
    const int stride = gridDim.x * blockDim.x;

    #pragma unroll 2
    for (int i = tid; i < n4; i += stride) {
        // gfx1250: HBM -> L2 prefetch two grid-strides ahead (192MB L2 holds the in-flight window)
        __builtin_prefetch(&x4[i + 2 * stride], 0, 1);

        v4f xv = __builtin_nontemporal_load(&x4[i]);   // global_load_b128 th:NT (read-once stream)
        float xin[4] = {xv.x, xv.y, xv.z, xv.w};
        float yo[4], lo[4];
        #pragma unroll
        for (int e = 0; e < 4; ++e) {
            float xx = xin[e];
            float xc = fminf(fmaxf(xx, -BOUNDF), BOUNDF);
            int cell = (int)fmaf(xc, cellScale, cellBias);
            cell = min(cell, LUTN - 1);
            int2 ce = sCell[cell];                        // ds_load_b64: {boundary, bin0}
            float nb = __int_as_float(ce.x);
            int bin = ce.y + ((nb < xc) ? 1 : 0);         // == searchsorted
            float4 p0 = sP0[bin];                         // ds_load_b128, bank-conflict-free
            float4 p1 = sP1[bin];
            float xi  = (xc - p0.x) * p0.y;
            float om  = 1.0f - xi;
            float t   = xi * om;
            float sk  = p1.z;
            float num = fmaf(sk * xi, xi, p1.x * t);
            float den = fmaf(p1.w, t, sk);
            float r   = __builtin_amdgcn_rcpf(den);       // v_rcp_f32 (TRANS)
            float yi  = fmaf(p0.w * num, r, p0.z);
            float dn  = fmaf(p1.y * xi, xi, fmaf(sk + sk, t, (p1.x * om) * om));
            float dv  = ((sk * sk) * dn) * (r * r);
            // C1 spline: at clamped ends the interior formula gives y=knot_y, deriv=d_edge,
            // so tails are branch-free: y += deriv*(x - xc) (zero for interior points).
            yo[e] = fmaf(dv, xx - xc, yi);
            lo[e] = __logf(fmaxf(dv, 1e-8f));             // v_log_f32 (TRANS)
        }
        v4f yv = {yo[0], yo[1], yo[2], yo[3]};
        v4f lv = {lo[0], lo[1], lo[2], lo[3]};
        __builtin_nontemporal_store(yv, &y4[i]);          // global_store_b128 th:NT (write-once stream)
        __builtin_nontemporal_store(lv, &l4[i]);
    }

    // scalar tail (n not divisible by 4; no-op for 8192x8192)
    for (int i = n4 * 4 + tid; i < n; i += stride) {
        float xx = x[i];
        float xc = fminf(fmaxf(xx, -BOUNDF), BOUNDF);
        int cell = min((int)fmaf(xc, cellScale, cellBias), LUTN - 1);
        int2 ce = sCell[cell];
        float nb = __int_as_float(ce.x);
        int bin = ce.y + ((nb < xc) ? 1 : 0);
        float4 p0 = sP0[bin];
        float4 p1 = sP1[bin];
        float xi  = (xc - p0.x) * p0.y;
        float om  = 1.0f - xi;
        float t   = xi * om;
        float sk  = p1.z;
        float num = fmaf(sk * xi, xi, p1.x * t);
        float den = fmaf(p1.w, t, sk);
        float r   = __builtin_amdgcn_rcpf(den);
        float yi  = fmaf(p0.w * num, r, p0.z);
        float dn  = fmaf(p1.y * xi, xi, fmaf(sk + sk, t, (p1.x * om) * om));
        float dv  = ((sk * sk) * dn) * (r * r);
        out_y[i]  = fmaf(dv, xx - xc, yi);
        out_ld[i] = __logf(fmaxf(dv, 1e-8f));
    }
}

extern "C" void kernel_launch(void* const* d_in, const int* in_sizes, int n_in,
                              void* d_out, int out_size, void* d_ws, size_t ws_size,
                              hipStream_t stream) {
    (void)n_in; (void)out_size; (void)d_ws; (void)ws_size;
    const float* x  = (const float*)d_in[0];
    const float* uw = (const float*)d_in[1];
    const float* uh = (const float*)d_in[2];
    const float* ud = (const float*)d_in[3];
    const int n = in_sizes[0];
    float* out_y  = (float*)d_out;
    float* out_ld = out_y + n;
    const int n4 = n / 4;

    const int threads = 256;            // 8 wave32 per block
    int blocks = 4096;                  // long-lived blocks amortize the per-block prologue
    int needed = (n4 + threads - 1) / threads;
    if (needed < 1) needed = 1;
    if (blocks > needed) blocks = needed;

    rqs_kernel<<<blocks, threads, 0, stream>>>(x, uw, uh, ud, out_y, out_ld, n4, n);
}